// VectorQuantizer_41661182771666
// MI455X (gfx1250) — compile-verified
//
#include <hip/hip_runtime.h>

typedef __attribute__((ext_vector_type(2))) float v2f;
typedef __attribute__((ext_vector_type(4))) float v4f;
typedef __attribute__((ext_vector_type(8))) float v8f;

#define NUM_EMBEDS 8192
#define EMBED_DIM  64
#define BATCH      16
#define HW         1024            // 32*32
#define NVEC       (BATCH * HW)    // 16384 latent vectors
#define NELEM      (BATCH * EMBED_DIM * HW)  // 1048576 output elements
#define CHUNK      128             // embeds staged in LDS per iteration
#define LDS_STRIDE 68              // 64 + 4 pad floats -> conflict-free ds reads

// ---------------------------------------------------------------- e_sq
__global__ __launch_bounds__(256) void vq_esq(const float* __restrict__ embed,
                                              float* __restrict__ esq) {
  int i = blockIdx.x * 256 + threadIdx.x;
  if (i >= NUM_EMBEDS) return;
  const v4f* row = (const v4f*)(embed + (size_t)i * EMBED_DIM);
  float s = 0.f;
#pragma unroll
  for (int j = 0; j < EMBED_DIM / 4; ++j) {
    v4f v = row[j];
    s += v.x * v.x + v.y * v.y + v.z * v.z + v.w * v.w;
  }
  esq[i] = s;
}

// ---------------------------------------------------------------- init
__global__ void vq_init(float* __restrict__ loss_acc) { loss_acc[0] = 0.f; }

// ---------------------------------------------------------------- argmin (WMMA fp32)
// Block = 256 threads (8 waves). Each wave owns a 16-row x tile; block owns 128 rows.
// Score tile D[M=x rows, N=embed cols] accumulated with 16 x V_WMMA_F32_16X16X4_F32.
__global__ __launch_bounds__(256) void vq_argmin(const float* __restrict__ x,
                                                 const float* __restrict__ embed,
                                                 const float* __restrict__ esq,
                                                 int* __restrict__ idx_out) {
  __shared__ float lds[CHUNK * LDS_STRIDE];

  const int tid  = threadIdx.x;
  const int wave = tid >> 5;
  const int lane = tid & 31;
  const int half = lane >> 4;   // 0: M=0..7 slots, 1: M=8..15 slots
  const int l16  = lane & 15;
  const int x_base = blockIdx.x * 128 + wave * 16;

  // --- load this wave's 16x64 x tile as 16 A fragments (16x4 fp32 each) ---
  // x layout is [B, C, H*W]; row m, component c lives at b*C*HW + c*HW + hw.
  const int m  = x_base + l16;
  const int bb = m >> 10;
  const int hw = m & 1023;
  const float* xrow = x + (size_t)bb * (EMBED_DIM * HW) + hw;
  v2f a[16];
#pragma unroll
  for (int kk = 0; kk < 16; ++kk) {
    const int k = kk * 4 + 2 * half;       // A lane layout: K = 2*(lane/16) + vgpr
    a[kk].x = xrow[(size_t)k * HW];
    a[kk].y = xrow[(size_t)(k + 1) * HW];
  }

  float min_s[8];
  int   min_i[8];
#pragma unroll
  for (int r = 0; r < 8; ++r) { min_s[r] = 3.4e38f; min_i[r] = 0; }

  for (int chunk = 0; chunk < NUM_EMBEDS / CHUNK; ++chunk) {
    const int eb = chunk * CHUNK;
    __syncthreads();
    // cooperative stage: 128 embed rows -> LDS with padded stride
#pragma unroll
    for (int i = 0; i < 8; ++i) {
      const int f4   = i * 256 + tid;       // 2048 float4 per chunk
      const int row  = f4 >> 4;             // 16 float4 per embed row
      const int col4 = f4 & 15;
      v4f v = *(const v4f*)(embed + (size_t)(eb + row) * EMBED_DIM + col4 * 4);
      *(v4f*)(&lds[row * LDS_STRIDE + col4 * 4]) = v;
    }
    __syncthreads();
    if (chunk + 1 < NUM_EMBEDS / CHUNK)     // warm GL2 for the next chunk
      __builtin_prefetch(embed + (size_t)(eb + CHUNK) * EMBED_DIM + tid * 32, 0, 1);

#pragma unroll
    for (int et = 0; et < CHUNK / 16; ++et) {
      // B fragment: lane holds B[k, n] with n = lane%16, k = 2*(lane/16)+vgpr
      const float* brow = &lds[(et * 16 + l16) * LDS_STRIDE + 2 * half];
      v8f acc = {};
#pragma unroll
      for (int kk = 0; kk < 16; ++kk) {
        v2f bfrag = *(const v2f*)(brow + kk * 4);
        acc = __builtin_amdgcn_wmma_f32_16x16x4_f32(
            /*neg_a=*/false, a[kk], /*neg_b=*/false, bfrag,
            /*c_mod=*/(short)0, acc, /*reuse_a=*/false, /*reuse_b=*/false);
      }
      const int   e_idx = eb + et * 16 + l16;
      const float es    = esq[e_idx];
#pragma unroll
      for (int r = 0; r < 8; ++r) {
        const float s = es - 2.f * acc[r];  // ||e||^2 - 2 x.e  (||x||^2 dropped)
        if (s < min_s[r]) { min_s[r] = s; min_i[r] = e_idx; }
      }
    }
  }

  // --- final argmin across the 16 lanes of each half-wave (N dimension) ---
#pragma unroll
  for (int r = 0; r < 8; ++r) {
    float s = min_s[r];
    int   i = min_i[r];
#pragma unroll
    for (int off = 8; off >= 1; off >>= 1) {
      const float os = __shfl_xor(s, off, 32);
      const int   oi = __shfl_xor(i, off, 32);
      if (os < s || (os == s && oi < i)) { s = os; i = oi; }
    }
    if (l16 == 0) idx_out[x_base + half * 8 + r] = i;  // row M = r + 8*half
  }
}

// ---------------------------------------------------------------- gather + loss
__global__ __launch_bounds__(256) void vq_gather(const float* __restrict__ x,
                                                 const float* __restrict__ embed,
                                                 const int* __restrict__ idx,
                                                 float* __restrict__ out,
                                                 float* __restrict__ loss_acc) {
  const int p   = blockIdx.x * 256 + threadIdx.x;  // flat [B, C, H*W] index
  const int b   = p >> 16;                         // C*HW = 65536
  const int rem = p & 65535;
  const int c   = rem >> 10;
  const int hw  = rem & 1023;
  const int n   = b * 1024 + hw;
  const float q  = embed[(size_t)idx[n] * EMBED_DIM + c];
  const float xv = x[p];
  out[p] = q;                                      // x_q_st forward value == x_q
  const float d = q - xv;
  float local = d * d;
#pragma unroll
  for (int off = 16; off >= 1; off >>= 1) local += __shfl_down(local, off, 32);
  if ((threadIdx.x & 31) == 0) atomicAdd(loss_acc, local);
}

__global__ void vq_finalize(const float* __restrict__ loss_acc,
                            float* __restrict__ loss_out) {
  // vq_loss = (1 + BETA) * mean((x_q - x)^2), BETA = 0.25
  loss_out[0] = loss_acc[0] * (1.25f / (float)NELEM);
}

// ---------------------------------------------------------------- launch
extern "C" void kernel_launch(void* const* d_in, const int* in_sizes, int n_in,
                              void* d_out, int out_size, void* d_ws, size_t ws_size,
                              hipStream_t stream) {
  (void)in_sizes; (void)n_in; (void)ws_size;
  const float* x     = (const float*)d_in[0];   // [16, 64, 32, 32] fp32
  const float* embed = (const float*)d_in[1];   // [8192, 64] fp32
  float* out = (float*)d_out;                   // [1048576 x_q_st] + [1 loss]

  float* esq      = (float*)d_ws;
  int*   idx      = (int*)((char*)d_ws + NUM_EMBEDS * sizeof(float));
  float* loss_acc = (float*)((char*)d_ws + NUM_EMBEDS * sizeof(float)
                                         + NVEC * sizeof(int));

  vq_init<<<1, 1, 0, stream>>>(loss_acc);
  vq_esq<<<NUM_EMBEDS / 256, 256, 0, stream>>>(embed, esq);
  vq_argmin<<<NVEC / 128, 256, 0, stream>>>(x, embed, esq, idx);
  vq_gather<<<NELEM / 256, 256, 0, stream>>>(x, embed, idx, out, loss_acc);
  vq_finalize<<<1, 1, 0, stream>>>(loss_acc, out + (out_size - 1));
}